// SigScaleLayer_87333864997628
// MI455X (gfx1250) — compile-verified
//
#include <hip/hip_runtime.h>
#include <cstdint>

typedef __attribute__((ext_vector_type(4))) float f4;

#define SIGLEN 299592u   // 8+64+512+4096+32768+262144
#define SIG4   74898u    // SIGLEN/4
#define TPB    256
#define ITERS  8

// Level element-offset boundaries (element units, all multiples of 4):
//   L1:[0,8) L2:[8,72) L3:[72,584) L4:[584,4680) L5:[4680,37448) L6:[37448,299592)

__global__ __launch_bounds__(TPB) void sigscale_tdm_kernel(
    const float* __restrict__ x, const float* __restrict__ y,
    float* __restrict__ out)
{
    // Prefix-factor tables: g1 = y, g2 = y(x)y, g3 = y(x)y(x)y
    __shared__ alignas(16) float g1[8];
    __shared__ alignas(16) float g2[64];
    __shared__ alignas(16) float g3[512];
    // Double-buffered async staging: 2 x 256 lanes x 16B = 8KB
    __shared__ alignas(16) float stage[2][TPB * 4];

    const unsigned tid = threadIdx.x;
    const unsigned b   = blockIdx.y;
    const float* __restrict__ xrow = x   + (size_t)b * SIGLEN;
    const float* __restrict__ yrow = y   + (size_t)b * 8u;
    float*       __restrict__ orow = out + (size_t)b * SIGLEN;

    // ---- build factor tables in LDS ----
    if (tid < 8u)  g1[tid] = yrow[tid];
    __syncthreads();
    if (tid < 64u) g2[tid] = g1[tid >> 3] * g1[tid & 7u];
    __syncthreads();
    {
        const unsigned t0 = tid, t1 = tid + 256u;
        g3[t0] = g2[t0 >> 3] * g1[t0 & 7u];
        g3[t1] = g2[t1 >> 3] * g1[t1 & 7u];
    }
    __syncthreads();

    const unsigned p0 = blockIdx.x * (TPB * ITERS) + tid;   // float4 index in row
    // AS3 offset of `stage`: flat LDS addr = {aperture_hi32, lds_offset32}
    const unsigned stage_off = (unsigned)(size_t)(void*)&stage[0][0];

    // ---- prologue: issue async DMA of tile 0 into LDS buffer 0 (NT: no reuse) ----
    {
        const unsigned pc = (p0 < SIG4) ? p0 : (SIG4 - 1u);
        const float*   gp  = xrow + (size_t)pc * 4u;
        const unsigned lds = stage_off + tid * 16u;
        asm volatile("global_load_async_to_lds_b128 %0, %1, off th:TH_LOAD_NT"
                     :: "v"(lds), "v"(gp) : "memory");
    }

    #pragma unroll
    for (int j = 0; j < ITERS; ++j) {
        if (j + 1 < ITERS) {
            // Prefetch next tile into the other buffer, then wait for current.
            const unsigned pn  = p0 + (unsigned)(j + 1) * TPB;
            const unsigned pnc = (pn < SIG4) ? pn : (SIG4 - 1u);
            const float*   gp  = xrow + (size_t)pnc * 4u;
            const unsigned lds = stage_off + (unsigned)((j + 1) & 1) * (TPB * 16u)
                               + tid * 16u;
            asm volatile("s_wait_dscnt 0x0\n\t"                 // recycled slot fully read
                         "global_load_async_to_lds_b128 %0, %1, off th:TH_LOAD_NT"
                         :: "v"(lds), "v"(gp) : "memory");
            asm volatile("s_wait_asynccnt 0x1" ::: "memory");   // tile j landed
        } else {
            asm volatile("s_wait_asynccnt 0x0" ::: "memory");   // last tile landed
        }

        const unsigned p     = p0 + (unsigned)j * TPB;
        const bool     valid = p < SIG4;
        const unsigned pc    = valid ? p : (SIG4 - 1u);
        const unsigned e     = pc * 4u;          // element index in row, e % 4 == 0

        const f4 xv = *reinterpret_cast<const f4*>(&stage[j & 1][tid * 4u]);

        // factor for 4 consecutive elements = hi * LO[lo .. lo+3]
        float hi = 1.0f;
        const float* LO; unsigned lo;
        if (e < 584u) {
            if (e < 8u)       { LO = g1; lo = e;        }
            else if (e < 72u) { LO = g2; lo = e - 8u;   }
            else              { LO = g3; lo = e - 72u;  }
        } else if (e < 4680u) {
            const unsigned n = e - 584u;   hi = g3[n >> 3]; LO = g1; lo = n & 7u;
        } else if (e < 37448u) {
            const unsigned n = e - 4680u;  hi = g3[n >> 6]; LO = g2; lo = n & 63u;
        } else {
            const unsigned n = e - 37448u; hi = g3[n >> 9]; LO = g3; lo = n & 511u;
        }

        const f4 lov = *reinterpret_cast<const f4*>(LO + lo);  // ds_load_b128 (16B aligned)
        const f4 r   = xv * (lov * hi);

        if (valid)  // streaming output: 128-bit coalesced, non-temporal
            __builtin_nontemporal_store(r, reinterpret_cast<f4*>(orow + e));
    }
}

extern "C" void kernel_launch(void* const* d_in, const int* in_sizes, int n_in,
                              void* d_out, int out_size, void* d_ws, size_t ws_size,
                              hipStream_t stream) {
    const float* x = (const float*)d_in[0];
    const float* y = (const float*)d_in[1];
    float* out = (float*)d_out;
    const int B = in_sizes[1] / 8;               // 256
    dim3 grid((SIG4 + TPB * ITERS - 1u) / (TPB * ITERS), (unsigned)B);
    sigscale_tdm_kernel<<<grid, TPB, 0, stream>>>(x, y, out);
}